// NeuralCollapseLoss_23201413333225
// MI455X (gfx1250) — compile-verified
//
#include <hip/hip_runtime.h>
#include <math.h>

// NeuralCollapseLoss on MI455X (gfx1250, wave32).
// dist^2 per 16-sample tile = diagonal of Gram(diff) computed with
// V_WMMA_F32_16X16X4_F32, accumulating over D=512 in K=4 steps.
// Memory-bound: 512 MB feature stream (non-temporal), means gather served
// from L2 (2 MB table), ~22 us floor at 23.3 TB/s.

typedef __attribute__((ext_vector_type(2))) float v2f;
typedef __attribute__((ext_vector_type(8))) float v8f;

#define NCL_B 262144
#define NCL_D 512
#define NCL_EPS 5.0f

__global__ void __launch_bounds__(64) ncl_zero_kernel(float* out) {
    if (threadIdx.x == 0) out[0] = 0.0f;
}

__global__ void __launch_bounds__(256) ncl_main_kernel(
    const float* __restrict__ features,   // [B, 512]
    const float* __restrict__ means,      // [1000, 512]
    const int*   __restrict__ labels,     // [B]
    float* __restrict__ out)              // [1]
{
    const int lane = threadIdx.x & 31;
    const int wave = threadIdx.x >> 5;
    const int tile = blockIdx.x * 8 + wave;          // 16 samples per wave

    const int s    = lane & 15;                      // sample within tile
    const int half = lane >> 4;                      // K-pair select (0 -> K{0,1}, 1 -> K{2,3})

    const long long row = (long long)tile * 16 + s;
    const int cls = labels[row];

    const float* fp = features + row * (long long)NCL_D + half * 2;
    const float* mp = means + (long long)cls * NCL_D + half * 2;

    v8f acc = {};

    // D=512 -> 128 K-steps of 4. Per step each lane supplies a float2 of
    // diff; the same register pair is a valid A (16x4) AND B (4x16) operand
    // for the f32 WMMA (layouts coincide for Gram products).
#pragma unroll 8
    for (int kb = 0; kb < NCL_D / 4; ++kb) {
        v2f fv = __builtin_nontemporal_load((const v2f*)(fp + kb * 4)); // stream, don't cache
        v2f mv = *(const v2f*)(mp + kb * 4);                             // L2-resident gather
        v2f ab = fv - mv;
        // D = A*B + C  (A = diff tile 16x4, B = diff^T 4x16, C = Gram acc)
        acc = __builtin_amdgcn_wmma_f32_16x16x4_f32(
            /*neg_a=*/false, ab, /*neg_b=*/false, ab,
            /*c_mod=*/(short)0, acc, /*reuse_a=*/false, /*reuse_b=*/false);
    }

    // Diagonal of the 16x16 f32 C/D layout:
    //   lanes 0-15 : VGPR r holds (M=r,   N=lane)   -> diag at lanes 0-7, acc[lane]
    //   lanes 16-31: VGPR r holds (M=r+8, N=lane-16)-> diag at lanes 24-31, acc[lane-24]
    const bool active = (lane < 8) || (lane >= 24);
    int sel = ((lane < 8) ? lane : (lane - 24)) & 7;

    float d2 = acc[0];
#pragma unroll
    for (int r = 1; r < 8; ++r) {
        if (sel == r) d2 = acc[r];
    }

    float contrib = 0.0f;
    if (active) {
        float dist = sqrtf(d2);
        float h = NCL_EPS - dist;
        contrib = (h > 0.0f) ? h : 0.0f;
    }

    // Wave32 reduction (16 active lanes carry one sample each).
#pragma unroll
    for (int off = 16; off > 0; off >>= 1) {
        contrib += __shfl_xor(contrib, off, 32);
    }

    if (lane == 0) {
        atomicAdd(out, contrib * (1.0f / (float)NCL_B));
    }
}

extern "C" void kernel_launch(void* const* d_in, const int* in_sizes, int n_in,
                              void* d_out, int out_size, void* d_ws, size_t ws_size,
                              hipStream_t stream) {
    const float* features = (const float*)d_in[0];
    const float* means    = (const float*)d_in[1];
    const int*   labels   = (const int*)d_in[2];
    float* out = (float*)d_out;

    // d_out is poisoned by the harness: zero it first (stream-ordered).
    ncl_zero_kernel<<<1, 64, 0, stream>>>(out);

    // 16384 tiles of 16 samples, 8 waves per block -> 2048 blocks.
    ncl_main_kernel<<<NCL_B / 16 / 8, 256, 0, stream>>>(features, means, labels, out);
}